// canODE_attentionNoValue_40922448396538
// MI455X (gfx1250) — compile-verified
//
#include <hip/hip_runtime.h>
#include <hip/hip_bf16.h>

typedef __attribute__((ext_vector_type(2))) float v2f;
typedef __attribute__((ext_vector_type(8))) float v8f;

#define BATCH 64
#define NN 512
#define DD 128
#define ROWS (BATCH * NN)     // 32768
#define EPAD 132              // LDS row pitch (floats): no bank conflicts, 8B aligned
#define WPROWS 129            // padded weight rows (row 128 == 0)
#define SCALE_C 0.08838834764831843f  // 128^-0.5

// ---------------------------------------------------------------------------
// Kernel 0: stage zero-padded weights into workspace so the WMMA inner loop
// has no boundary conditionals (no divergent exec branches in the hot loop).
//   wp[r][d] = (r < 128) ? w[r][d] : 0     for r in [0, 129)
// ---------------------------------------------------------------------------
__global__ __launch_bounds__(256) void pad_weights_kernel(
    const float* __restrict__ wq_w, const float* __restrict__ wk_w,
    float* __restrict__ wqp, float* __restrict__ wkp)
{
    const int i = blockIdx.x * 256 + threadIdx.x;   // [0, 129*128)
    if (i < WPROWS * DD) {
        const bool inb = i < DD * DD;
        wqp[i] = inb ? wq_w[i] : 0.0f;
        wkp[i] = inb ? wk_w[i] : 0.0f;
    }
}

// ---------------------------------------------------------------------------
// Kernel 1: q_const/k_const precompute via f32 WMMA.
//   q_const[r,:] = embed[pos[r]] @ wq_w[1:,:] + wq_b   (K padded 127 -> 128)
// One wave (32 threads) per 16-row tile; A tile staged in LDS; for each of the
// 8 column tiles we run 32x V_WMMA_F32_16X16X4_F32 for q and for k, reusing
// the same A fragment.
// Fragment layouts (ISA 7.12.2):
//   A 16x4:  lanes 0-15: M=lane, v0=K0,v1=K1 ; lanes 16-31: M=lane-16, v0=K2,v1=K3
//   B 4x16:  mirrored: lanes 0-15: N=lane, v0=K0,v1=K1 ; lanes 16-31: v0=K2,v1=K3
//   C/D:     vgpr r: lanes 0-15 -> (M=r, N=lane); lanes 16-31 -> (M=8+r, N=lane-16)
// ---------------------------------------------------------------------------
__global__ __launch_bounds__(32) void precompute_qk_kernel(
    const int* __restrict__ pos, const float* __restrict__ embed,
    const float* __restrict__ wqp, const float* __restrict__ wq_b,
    const float* __restrict__ wkp, const float* __restrict__ wk_b,
    float* __restrict__ qc, float* __restrict__ kc)
{
    __shared__ float As[16 * EPAD];
    const int lane = threadIdx.x;
    const int rowBase = blockIdx.x * 16;

    // Stage A tile: 16 rows x 128 K (K=127 zero pad). Each lane: 1/2 row.
    {
        const int lrow = lane >> 1;
        const int cbase = (lane & 1) * 64;
        const long p = (long)pos[rowBase + lrow] * 127;
        #pragma unroll 8
        for (int j = 0; j < 64; ++j) {
            const int k = cbase + j;
            As[lrow * EPAD + k] = (k < 127) ? embed[p + k] : 0.0f;
        }
    }
    __syncthreads();

    const int mlo  = lane & 15;
    const bool hi  = lane >= 16;
    const int koff = hi ? 2 : 0;

    for (int dt = 0; dt < 8; ++dt) {
        const int dcol = dt * 16 + mlo;
        const float qb = wq_b[dcol];
        const float kb = wk_b[dcol];
        v8f accq = {qb, qb, qb, qb, qb, qb, qb, qb};  // bias broadcast along M
        v8f acck = {kb, kb, kb, kb, kb, kb, kb, kb};

        #pragma unroll 4
        for (int k0 = 0; k0 < 128; k0 += 4) {
            const int kk = k0 + koff;              // even, so As read is 8B aligned
            v2f a;
            a[0] = As[mlo * EPAD + kk];
            a[1] = As[mlo * EPAD + kk + 1];
            // padded weights: rows 1..128 valid, row 128 == 0 -> no conditionals
            v2f bq, bk;
            bq[0] = wqp[(kk + 1) * DD + dcol];
            bq[1] = wqp[(kk + 2) * DD + dcol];
            bk[0] = wkp[(kk + 1) * DD + dcol];
            bk[1] = wkp[(kk + 2) * DD + dcol];

            accq = __builtin_amdgcn_wmma_f32_16x16x4_f32(
                false, a, false, bq, (short)0, accq, false, false);
            acck = __builtin_amdgcn_wmma_f32_16x16x4_f32(
                false, a, false, bk, (short)0, acck, false, false);
        }

        #pragma unroll
        for (int r = 0; r < 8; ++r) {
            const long grow = rowBase + r + (hi ? 8 : 0);
            qc[grow * DD + dcol] = accq[r];
            kc[grow * DD + dcol] = acck[r];
        }
    }
}

// ---------------------------------------------------------------------------
// Kernel 2: full RK4 integration, one block (512 threads, 16 waves) per batch.
// vf(x) = x .* (f - <x,f>) with
//   s   = K_const^T x + (sum x^2) * wk0
//   f_i = SCALE * (q_const_i . s + x_i * (wq0 . s))
// ---------------------------------------------------------------------------
#define NT 512

__device__ __forceinline__ float blockReduce(float v, float* red, float* bcast, int tid)
{
    #pragma unroll
    for (int off = 16; off > 0; off >>= 1) v += __shfl_xor(v, off, 32);
    if ((tid & 31) == 0) red[tid >> 5] = v;
    __syncthreads();
    if (tid < 32) {
        float r = (tid < NT / 32) ? red[tid] : 0.0f;
        #pragma unroll
        for (int off = 16; off > 0; off >>= 1) r += __shfl_xor(r, off, 32);
        if (tid == 0) *bcast = r;
    }
    __syncthreads();
    return *bcast;
}

__device__ __forceinline__ void vf_eval(
    const float* __restrict__ qcb, const float* __restrict__ kcb,
    const float* xt, float* dst, float* scratch, float* s,
    const float* wq0, const float* wk0, float* red, float* scal, int tid)
{
    const float xi = xt[tid];
    // sum x^2
    const float sx2 = blockReduce(xi * xi, red, &scal[0], tid);

    // t[d] = sum_j kc[j][d] * x[j]   (4 row-quarters of 128 per d)
    const int d = tid & 127;
    const int q = tid >> 7;          // 0..3
    float part = 0.0f;
    const float* kcol = kcb + (long)q * 128 * DD + d;
    const float* xq = xt + q * 128;
    #pragma unroll 4
    for (int j = 0; j < 128; ++j) part += kcol[(long)j * DD] * xq[j];
    scratch[tid] = part;
    __syncthreads();
    if (tid < DD)
        s[tid] = scratch[tid] + scratch[tid + 128] + scratch[tid + 256] +
                 scratch[tid + 384] + sx2 * wk0[tid];
    __syncthreads();

    // c0 = wq0 . s
    const float c0 = blockReduce((tid < DD) ? wq0[tid] * s[tid] : 0.0f,
                                 red, &scal[1], tid);

    // f_i = SCALE * (q_const_i . s + x_i * c0)
    const float4* qrow = (const float4*)(qcb + (long)tid * DD);
    float dotv = 0.0f;
    #pragma unroll 8
    for (int d4 = 0; d4 < DD / 4; ++d4) {
        const float4 qv = qrow[d4];
        dotv += qv.x * s[4 * d4] + qv.y * s[4 * d4 + 1] +
                qv.z * s[4 * d4 + 2] + qv.w * s[4 * d4 + 3];
    }
    const float fi = SCALE_C * (dotv + xi * c0);

    // g = <x, f> ; vf = x .* (f - g)
    const float g = blockReduce(xi * fi, red, &scal[2], tid);
    dst[tid] = xi * (fi - g);
    __syncthreads();
}

__global__ __launch_bounds__(NT) void dynamics_kernel(
    const float* __restrict__ tg, const float* __restrict__ val,
    const float* __restrict__ wq_w, const float* __restrict__ wk_w,
    const float* __restrict__ qc, const float* __restrict__ kc,
    float* __restrict__ out)
{
    const int b = blockIdx.x;
    const int tid = threadIdx.x;

    __shared__ float x[NN], xt[NN], scratch[NN];
    __shared__ float kb1[NN], kb2[NN], kb3[NN], kb4[NN];
    __shared__ float s[DD], wq0[DD], wk0[DD];
    __shared__ float red[NT / 32];
    __shared__ float scal[4];

    const float* qcb = qc + (long)b * NN * DD;
    const float* kcb = kc + (long)b * NN * DD;

    const float x0 = val[b * NN + tid];
    x[tid] = x0;
    out[(long)b * NN + tid] = x0;                 // out[0] = val
    if (tid < DD) { wq0[tid] = wq_w[tid]; wk0[tid] = wk_w[tid]; }  // row 0 = x column
    __syncthreads();

    for (int step = 0; step < 9; ++step) {
        const float dt = tg[step + 1] - tg[step];

        vf_eval(qcb, kcb, x, kb1, scratch, s, wq0, wk0, red, scal, tid);
        xt[tid] = x[tid] + 0.5f * dt * kb1[tid];
        __syncthreads();
        vf_eval(qcb, kcb, xt, kb2, scratch, s, wq0, wk0, red, scal, tid);
        xt[tid] = x[tid] + 0.5f * dt * kb2[tid];
        __syncthreads();
        vf_eval(qcb, kcb, xt, kb3, scratch, s, wq0, wk0, red, scal, tid);
        xt[tid] = x[tid] + dt * kb3[tid];
        __syncthreads();
        vf_eval(qcb, kcb, xt, kb4, scratch, s, wq0, wk0, red, scal, tid);

        const float xn = x[tid] + (dt * (1.0f / 6.0f)) *
                         (kb1[tid] + 2.0f * kb2[tid] + 2.0f * kb3[tid] + kb4[tid]);
        __syncthreads();
        x[tid] = xn;
        out[(long)(step + 1) * (BATCH * NN) + (long)b * NN + tid] = xn;
        __syncthreads();
    }
}

extern "C" void kernel_launch(void* const* d_in, const int* in_sizes, int n_in,
                              void* d_out, int out_size, void* d_ws, size_t ws_size,
                              hipStream_t stream)
{
    (void)in_sizes; (void)n_in; (void)out_size; (void)ws_size;
    const float* t      = (const float*)d_in[0];   // [10]
    const float* val    = (const float*)d_in[1];   // [64*512]
    const int*   pos    = (const int*)  d_in[2];   // [64*512]
    const float* embed  = (const float*)d_in[3];   // [10001*127]
    const float* wq_w   = (const float*)d_in[4];   // [128*128]
    const float* wq_b   = (const float*)d_in[5];   // [128]
    const float* wk_w   = (const float*)d_in[6];   // [128*128]
    const float* wk_b   = (const float*)d_in[7];   // [128]
    float* out = (float*)d_out;                    // [10*64*512]

    float* qc  = (float*)d_ws;                     // [32768*128] = 16 MB
    float* kc  = qc + (long)ROWS * DD;             // [32768*128] = 16 MB
    float* wqp = kc + (long)ROWS * DD;             // [129*128]
    float* wkp = wqp + WPROWS * DD;                // [129*128]

    pad_weights_kernel<<<(WPROWS * DD + 255) / 256, 256, 0, stream>>>(
        wq_w, wk_w, wqp, wkp);
    precompute_qk_kernel<<<ROWS / 16, 32, 0, stream>>>(
        pos, embed, wqp, wq_b, wkp, wk_b, qc, kc);
    dynamics_kernel<<<BATCH, NT, 0, stream>>>(
        t, val, wq_w, wk_w, qc, kc, out);
}